// ChannelDepsModule_61100204753650
// MI455X (gfx1250) — compile-verified
//
#include <hip/hip_runtime.h>

typedef __attribute__((ext_vector_type(2))) float v2f;
typedef __attribute__((ext_vector_type(8))) float v8f;

#define C_CH 192
#define NBLK 12      // 192 / 16 channel blocks
#define HW   4096    // 64*64 spatial stride between channels
#define NPIX 16      // pixels per tile (WMMA N dim)

__global__ __launch_bounds__(32)
void channel_deps_kernel(const float* __restrict__ x,
                         const float* __restrict__ mix,
                         const float* __restrict__ Wp,   // [191][192] row-major
                         const float* __restrict__ bp,   // [191]
                         float* __restrict__ out_y,
                         float* __restrict__ out_m)
{
    __shared__ float ybar_s[C_CH * NPIX];  // decoded channels, [c][p]
    __shared__ float acc_s[16 * NPIX];     // GEMM partials staged for diag step
    __shared__ float xs[16 * NPIX];        // x block
    __shared__ float ms[16 * NPIX];        // mix block
    __shared__ float ws[16 * 16];          // diagonal W block [m][j]

    const int lane = threadIdx.x;          // 0..31 (wave32)
    const int p    = lane & 15;            // pixel within tile
    const int half = lane >> 4;            // half-wave id (WMMA fragment striping)

    // pixel tile: 16 consecutive w, fixed (n, h)
    const int tile = blockIdx.x;           // 0..2047
    const int w0   = (tile & 3) << 4;
    const int h    = (tile >> 2) & 63;
    const int n    = tile >> 8;
    const size_t base = ((size_t)(n * C_CH) * 64 + h) * 64 + w0;

    for (int J = 0; J < NBLK; ++J) {
        v8f acc = {0.f, 0.f, 0.f, 0.f, 0.f, 0.f, 0.f, 0.f};

        // ---- off-diagonal: acc += W[J,K] @ ybar[K]   (V_WMMA_F32_16X16X4_F32)
        const int rowc = J * 16 + p;       // output channel of this A-fragment row
        for (int K = 0; K < J; ++K) {
            if (K + 1 < J)  // warm L2/L1 for next W panel (global_prefetch_b8)
                __builtin_prefetch(Wp + (size_t)(rowc - 1) * C_CH + (K + 1) * 16, 0, 0);
#pragma unroll
            for (int kk = 0; kk < 4; ++kk) {
                const int k0 = K * 16 + kk * 4;
                // A: 16x4 f32. lanes 0-15 -> cols k0,k0+1 ; lanes 16-31 -> k0+2,k0+3
                const float* ap = Wp + (size_t)(rowc - 1) * C_CH + k0 + 2 * half;
                v2f a;  a.x = ap[0];  a.y = ap[1];
                // B: 4x16 f32 from LDS ybar (row-per-half-VGPR striping)
                v2f bf;
                bf.x = ybar_s[(k0 + 2 * half + 0) * NPIX + p];
                bf.y = ybar_s[(k0 + 2 * half + 1) * NPIX + p];
                acc = __builtin_amdgcn_wmma_f32_16x16x4_f32(
                        false, a, false, bf, (short)0, acc, false, false);
            }
        }

        // ---- stage acc (C/D layout: VGPR g -> rows g / g+8) + x/mix/diag-W blocks
#pragma unroll
        for (int g = 0; g < 8; ++g)
            acc_s[(g + 8 * half) * NPIX + p] = acc[g];
#pragma unroll
        for (int i = 0; i < 8; ++i) {
            const int r = i * 2 + half;
            const int c = J * 16 + r;
            xs[r * NPIX + p] = x[base + (size_t)c * HW + p];
            ms[r * NPIX + p] = mix[base + (size_t)c * HW + p];
            if (c >= 1)
                ws[r * 16 + p] = Wp[(size_t)(c - 1) * C_CH + J * 16 + p];
        }
        __syncthreads();

        // ---- diagonal block: 16-step forward substitution, one pixel per lane
        if (lane < 16) {
            float yloc[16];  // intra-block ybar kept in registers (serial chain)
#pragma unroll
            for (int m = 0; m < 16; ++m) {
                const int   c  = J * 16 + m;
                const float xv = xs[m * NPIX + p];
                const float mv = ms[m * NPIX + p];
                float mc;
                if (J == 0 && m == 0) {
                    mc = mv;                       // channel 0: mix_out = mix
                } else {
                    float s = acc_s[m * NPIX + p];
#pragma unroll
                    for (int j = 0; j < m; ++j)    // uniform LDS broadcast of W
                        s = fmaf(ws[m * 16 + j], yloc[j], s);
                    mc = s + bp[c - 1] + mv;
                }
                const float yb = rintf(xv - mc) + mc;   // v_rndne: half-to-even
                yloc[m] = yb;
                ybar_s[c * NPIX + p]        = yb;
                out_y[base + (size_t)c * HW + p] = yb;
                out_m[base + (size_t)c * HW + p] = mc;
            }
        }
        __syncthreads();   // ybar_s visible to full wave for next J's B-fragments
    }
}

extern "C" void kernel_launch(void* const* d_in, const int* in_sizes, int n_in,
                              void* d_out, int out_size, void* d_ws, size_t ws_size,
                              hipStream_t stream) {
    const float* x   = (const float*)d_in[0];
    const float* mix = (const float*)d_in[1];
    const float* W   = (const float*)d_in[2];
    const float* b   = (const float*)d_in[3];
    float* out_y = (float*)d_out;
    float* out_m = (float*)d_out + (size_t)8 * C_CH * HW;   // ybar ++ mix_out

    dim3 grid(2048), block(32);   // one wave32 per 16-pixel tile
    hipLaunchKernelGGL(channel_deps_kernel, grid, block, 0, stream,
                       x, mix, W, b, out_y, out_m);
}